// GCNEncoder_26061861552480
// MI455X (gfx1250) — compile-verified
//
#include <hip/hip_runtime.h>

// ---------------- problem constants (match reference) ----------------
#define N_NODES 100000
#define N_EDGES 1600000
#define IN_DIM  128
#define HIDDEN  128
#define OUT_DIM 64

typedef __attribute__((ext_vector_type(2))) float v2f;
typedef __attribute__((ext_vector_type(8))) float v8f;

// ---------------- degree / normalization ----------------
__global__ void k_init_deg(float* __restrict__ deg) {
    int i = blockIdx.x * blockDim.x + threadIdx.x;
    if (i < N_NODES) deg[i] = 1.0f;   // self-loop contributes 1
}

__global__ void k_count_deg(const long long* __restrict__ ei, float* __restrict__ deg) {
    int e = blockIdx.x * blockDim.x + threadIdx.x;
    if (e < N_EDGES) {
        int dst = (int)ei[(long long)N_EDGES + e];
        atomicAdd(&deg[dst], 1.0f);
    }
}

__global__ void k_deg_to_rsqrt(float* __restrict__ deg) {
    int i = blockIdx.x * blockDim.x + threadIdx.x;
    if (i < N_NODES) deg[i] = rsqrtf(deg[i]);   // deg >= 1 always
}

// ---------------- fp32 WMMA GEMM: C[M,N] = A[M,K] * B[K,N] ----------------
// One wave computes one 16x16 tile of C. Block = (N/16) waves, one 16-row
// stripe of A per block. M (=N_NODES) is an exact multiple of 16, so EXEC is
// all-ones everywhere (WMMA requirement).
//
// Per ISA 7.12.2 (32-bit operands, wave32):
//   A 16x4 : lane -> M = lane&15 ; VGPR v -> K = kk + 2*(lane>>4) + v
//   B 4x16 : lane -> N = lane&15 ; VGPR v -> K = kk + 2*(lane>>4) + v
//   C 16x16: lane -> N = lane&15 ; VGPR r -> M = r + 8*(lane>>4)
template <int K, int N>
__global__ __launch_bounds__(256) void k_gemm_wmma_f32(
    const float* __restrict__ A, const float* __restrict__ B,
    float* __restrict__ C)
{
    const int wave = threadIdx.x >> 5;
    const int lane = threadIdx.x & 31;
    const int m0   = blockIdx.x * 16;
    const int n0   = wave * 16;
    const int ml   = lane & 15;
    const int half = lane >> 4;

    const float* __restrict__ Arow = A + (long long)(m0 + ml) * K;

    v8f acc = {};
#pragma unroll 4
    for (int kk = 0; kk < K; kk += 4) {
        const int ka = kk + 2 * half;
        v2f a, b;
        a.x = Arow[ka];
        a.y = Arow[ka + 1];
        b.x = B[(long long)ka * N + n0 + ml];
        b.y = B[(long long)(ka + 1) * N + n0 + ml];
        acc = __builtin_amdgcn_wmma_f32_16x16x4_f32(
            /*neg_a=*/false, a, /*neg_b=*/false, b,
            /*c_mod=*/(short)0, acc, /*reuse_a=*/false, /*reuse_b=*/false);
    }

#pragma unroll
    for (int r = 0; r < 8; ++r) {
        C[(long long)(m0 + r + 8 * half) * N + n0 + ml] = acc[r];
    }
}

// ---------------- self-loop + bias initializer ----------------
// out[i,f] = dinv[i]^2 * h[i,f] + bias[f]
template <int F>
__global__ void k_self_init(const float* __restrict__ h,
                            const float* __restrict__ dinv,
                            const float* __restrict__ bias,
                            float* __restrict__ out)
{
    long long idx = (long long)blockIdx.x * blockDim.x + threadIdx.x;
    if (idx < (long long)N_NODES * F) {
        int i = (int)(idx / F);
        int f = (int)(idx % F);
        float d = dinv[i];
        out[idx] = d * d * h[idx] + bias[f];
    }
}

// ---------------- edge scatter: out[dst] += dinv[src]*dinv[dst]*h[src] ----------------
// One wave per edge; lanes stride the feature dim -> coalesced 128B loads of
// h[src] and coalesced GLOBAL_ATOMIC_ADD_F32 bursts into out[dst].
template <int F>
__global__ __launch_bounds__(256) void k_scatter(
    const long long* __restrict__ ei,
    const float* __restrict__ h,
    const float* __restrict__ dinv,
    float* __restrict__ out)
{
    const int wave = threadIdx.x >> 5;
    const int lane = threadIdx.x & 31;
    const long long e = (long long)blockIdx.x * (blockDim.x >> 5) + wave;
    if (e >= N_EDGES) return;

    const int src = (int)ei[e];
    const int dst = (int)ei[(long long)N_EDGES + e];
    const float norm = dinv[src] * dinv[dst];

    const float* __restrict__ hs = h + (long long)src * F;
    float* __restrict__ od = out + (long long)dst * F;
#pragma unroll
    for (int j = 0; j < F / 32; ++j) {
        const int f = lane + 32 * j;
        atomicAdd(&od[f], norm * hs[f]);
    }
}

// ---------------- ReLU in place ----------------
__global__ void k_relu(float* __restrict__ x, long long total) {
    long long idx = (long long)blockIdx.x * blockDim.x + threadIdx.x;
    if (idx < total) x[idx] = fmaxf(x[idx], 0.0f);
}

// ---------------- launch ----------------
extern "C" void kernel_launch(void* const* d_in, const int* in_sizes, int n_in,
                              void* d_out, int out_size, void* d_ws, size_t ws_size,
                              hipStream_t stream) {
    (void)in_sizes; (void)n_in; (void)out_size; (void)ws_size;

    const float*     x  = (const float*)d_in[0];
    const long long* ei = (const long long*)d_in[1];   // int64 edge_index [2,E]
    const float*     W1 = (const float*)d_in[2];
    const float*     b1 = (const float*)d_in[3];
    const float*     W2 = (const float*)d_in[4];
    const float*     b2 = (const float*)d_in[5];
    float*           out = (float*)d_out;              // [N, OUT_DIM]

    // workspace layout (floats): dinv | h1 | agg1 | h2   (~128.4 MB total)
    float* dinv = (float*)d_ws;                              // N
    float* h1   = dinv + N_NODES;                            // N*HIDDEN
    float* agg1 = h1   + (long long)N_NODES * HIDDEN;        // N*HIDDEN
    float* h2   = agg1 + (long long)N_NODES * HIDDEN;        // N*OUT_DIM

    const int TPB = 256;

    // D^{-1/2} with self-loops (in-degree over dst)
    k_init_deg   <<<(N_NODES + TPB - 1) / TPB, TPB, 0, stream>>>(dinv);
    k_count_deg  <<<(N_EDGES + TPB - 1) / TPB, TPB, 0, stream>>>(ei, dinv);
    k_deg_to_rsqrt<<<(N_NODES + TPB - 1) / TPB, TPB, 0, stream>>>(dinv);

    // ---- layer 1: h1 = x @ W1 ; agg1 = Ahat*h1 + b1 ; relu ----
    k_gemm_wmma_f32<IN_DIM, HIDDEN>
        <<<N_NODES / 16, (HIDDEN / 16) * 32, 0, stream>>>(x, W1, h1);

    const long long t1 = (long long)N_NODES * HIDDEN;
    k_self_init<HIDDEN><<<(unsigned)((t1 + TPB - 1) / TPB), TPB, 0, stream>>>(h1, dinv, b1, agg1);
    k_scatter<HIDDEN><<<(N_EDGES + 7) / 8, TPB, 0, stream>>>(ei, h1, dinv, agg1);
    k_relu<<<(unsigned)((t1 + TPB - 1) / TPB), TPB, 0, stream>>>(agg1, t1);

    // ---- layer 2: h2 = agg1 @ W2 ; out = Ahat*h2 + b2 ----
    k_gemm_wmma_f32<HIDDEN, OUT_DIM>
        <<<N_NODES / 16, (OUT_DIM / 16) * 32, 0, stream>>>(agg1, W2, h2);

    const long long t2 = (long long)N_NODES * OUT_DIM;
    k_self_init<OUT_DIM><<<(unsigned)((t2 + TPB - 1) / TPB), TPB, 0, stream>>>(h2, dinv, b2, out);
    k_scatter<OUT_DIM><<<(N_EDGES + 7) / 8, TPB, 0, stream>>>(ei, h2, dinv, out);
}